// OptimizedSTGCN_PRISM_8993661517878
// MI455X (gfx1250) — compile-verified
//
#include <hip/hip_runtime.h>
#include <hip/hip_bf16.h>
#include <stdint.h>

typedef float v2f __attribute__((ext_vector_type(2)));
typedef float v8f __attribute__((ext_vector_type(8)));

// ---------------- problem constants ----------------
#define JNT   25
#define CIN   3
#define HCH   64
#define TLEN  1024
#define BAT   32
#define NROW  (BAT*JNT)                 // 800
#define NELEM ((unsigned)BAT*JNT*HCH*TLEN) // 52,428,800
#define BN_N  ((float)(NROW*TLEN))      // 819200
#define EPSV  1e-5f

// skeleton adjacency (links + self), CSR form
__constant__ int c_off[26] = {0,5,8,11,14,16,19,22,25,27,30,33,36,38,43,46,49,51,54,57,60,62,65,68,71,73};
__constant__ int c_nbr[73] = {
  0,1,5,9,13,
  0,1,2,  1,2,3,  2,3,4,  3,4,
  0,5,6,  5,6,7,  6,7,8,  7,8,
  0,9,10, 9,10,11, 10,11,12, 11,12,
  0,13,14,17,21,
  13,14,15, 14,15,16, 15,16,
  13,17,18, 17,18,19, 18,19,20, 19,20,
  13,21,22, 21,22,23, 22,23,24, 23,24
};

// ---------------- 1) input projection with the torch view quirk ----------------
// h[b,j,hc,t] = b_proj[t&63] + sum_c flat[r,c]*w_proj[t&63,c]
// r = (b*25+j)*1024 + hc*16 + (t>>6); flat element e=r*3+c decoded in (B,C,T,J) layout.
__global__ __launch_bounds__(256) void proj_kernel(const float* __restrict__ x,
                                                   const float* __restrict__ w_proj,
                                                   const float* __restrict__ b_proj,
                                                   float* __restrict__ h) {
  unsigned id = blockIdx.x * 256u + threadIdx.x;
  if (id >= NELEM) return;
  int t  = id & (TLEN - 1);
  int hc = (id >> 10) & 63;
  unsigned bj = id >> 16;                 // b*25+j
  unsigned r  = bj * 1024u + (unsigned)(hc * 16) + (unsigned)(t >> 6);
  int o = t & 63;
  unsigned e0 = r * 3u;
  float acc = b_proj[o];
#pragma unroll
  for (int c = 0; c < 3; ++c) {
    unsigned e  = e0 + (unsigned)c;
    unsigned j2 = e % 25u;
    unsigned q  = e / 25u;
    unsigned t2 = q & 1023u;
    unsigned q2 = q >> 10;
    unsigned c2 = q2 % 3u;
    unsigned b2 = q2 / 3u;
    float xv = x[((b2 * 1024u + t2) * 25u + j2) * 3u + c2];
    acc += xv * w_proj[o * 3 + c];
  }
  h[id] = acc;
}

// ---------------- 2) fold spatial linear into temporal conv weights ----------------
// weff[k][o][c] = sum_i w_tc[blk][o][i][k] * w_sp[blk][i][c]
__global__ __launch_bounds__(256) void weff_kernel(const float* __restrict__ w_sp,
                                                   const float* __restrict__ w_tc,
                                                   float* __restrict__ weff, int blk) {
  int id = blockIdx.x * 256 + threadIdx.x;       // 12288 total
  if (id >= 3 * 64 * 64) return;
  int k = id >> 12;
  int o = (id >> 6) & 63;
  int c = id & 63;
  float acc = 0.0f;
  for (int i = 0; i < 64; ++i)
    acc += w_tc[((blk * 64 + o) * 64 + i) * 3 + k] * w_sp[(blk * 64 + i) * 64 + c];
  weff[id] = acc;
}

__global__ void zero_kernel(float* p, int n) {
  int i = blockIdx.x * blockDim.x + threadIdx.x;
  if (i < n) p[i] = 0.0f;
}

// ---------------- 3) fused skeleton-agg + WMMA temporal conv + BN partial sums ----------
// Workgroup: one (row n=b*25+i, 128-wide t block). LDS staging computes
//   lh[c][tl] = sum_{j in nbr(i)} h[b,j,c,tbase+tl-1]   (adjacency fused into the load)
// then 8 waves compute u[o,t] = sum_{k,c} weff[k][o][c]*lh[c][t+k-1]
// via V_WMMA_F32_16X16X4_F32 (fp32 matrix path, keeps reference numerics).
__global__ __launch_bounds__(256) void conv_wmma_kernel(const float* __restrict__ h,
                                                        const float* __restrict__ weff,
                                                        float* __restrict__ u,
                                                        float* __restrict__ bnsum,
                                                        float* __restrict__ bnsq) {
  __shared__ float lh[64 * 130];          // [c][t_local], t_local = t - tbase + 1
  __shared__ float lbs[64], lbq[64];
  int tid  = threadIdx.x;
  int nrow = blockIdx.x >> 3;             // b*25 + i
  int tbase = (blockIdx.x & 7) * 128;
  if (tid < 64) { lbs[tid] = 0.0f; lbq[tid] = 0.0f; }

  int i  = nrow % 25;                     // joint (uniform per workgroup)
  const float* hb = h + (size_t)(nrow - i) * (HCH * TLEN);   // batch-row base
  int e0 = c_off[i], e1 = c_off[i + 1];
  for (int idx = tid; idx < 64 * 130; idx += 256) {
    int c = idx / 130;
    int tl = idx - c * 130;
    int tg = tbase + tl - 1;
    float acc = 0.0f;
    if (tg >= 0 && tg < TLEN) {           // SAME zero pad
      int off = c * TLEN + tg;
      for (int e = e0; e < e1; ++e)
        acc += hb[c_nbr[e] * (HCH * TLEN) + off];
    }
    lh[idx] = acc;
  }
  __syncthreads();

  int wave = tid >> 5, lane = tid & 31;
  int o_tile = wave & 3;                  // 16 output channels each
  int t64    = wave >> 2;                 // 64 t-columns each
  int m  = lane & 15;                     // M for A, N for B, col for C/D
  int kb = (lane >> 4) << 1;              // K sub-offset per half-wave

  v8f acc[4] = {};                        // 4 t16-tiles of 16x16 fp32
  for (int k = 0; k < 3; ++k) {
    for (int cs = 0; cs < 16; ++cs) {     // K loop over channels, 4 at a time
      int c0 = (cs << 2) + kb;
      // A fragment: weff[k][o_tile*16+m][c0 .. c0+1]
      v2f a = *(const v2f*)(weff + (((k << 6) + (o_tile << 4) + m) << 6) + c0);
#pragma unroll
      for (int tt = 0; tt < 4; ++tt) {
        int tib = (t64 << 6) + (tt << 4) + m + k;   // t_in_block + (k-1) + halo(+1)
        v2f bf;
        bf.x = lh[c0 * 130 + tib];
        bf.y = lh[(c0 + 1) * 130 + tib];
        acc[tt] = __builtin_amdgcn_wmma_f32_16x16x4_f32(
            false, a, false, bf, (short)0, acc[tt], false, false);
      }
    }
  }

  // write u + BatchNorm partial sums (sum, sum of squares per channel)
  float* urow = u + (size_t)nrow * (HCH * TLEN);
  int ohalf = (lane >> 4) << 3;
  for (int tt = 0; tt < 4; ++tt) {
    int tcol = tbase + (t64 << 6) + (tt << 4) + m;
#pragma unroll
    for (int r = 0; r < 8; ++r) {
      int o = (o_tile << 4) + ohalf + r;
      float v = acc[tt][r];
      urow[o * TLEN + tcol] = v;
      float s1 = v, s2 = v * v;
      for (int sh = 8; sh >= 1; sh >>= 1) {         // reduce over 16 t-columns
        s1 += __shfl_xor(s1, sh, 16);
        s2 += __shfl_xor(s2, sh, 16);
      }
      if (m == 0) { atomicAdd(&lbs[o], s1); atomicAdd(&lbq[o], s2); }
    }
  }
  __syncthreads();
  if (tid < 64) { atomicAdd(&bnsum[tid], lbs[tid]); atomicAdd(&bnsq[tid], lbq[tid]); }
}

// ---------------- 4) BN(training) + ReLU + residual into h ----------------
__global__ __launch_bounds__(256) void bn_apply_kernel(const float* __restrict__ u,
                                                       const float* __restrict__ bnsum,
                                                       const float* __restrict__ bnsq,
                                                       const float* __restrict__ gamma,
                                                       const float* __restrict__ beta,
                                                       float* __restrict__ h) {
  unsigned id = blockIdx.x * 256u + threadIdx.x;
  if (id >= NELEM) return;
  int o = (id >> 10) & 63;
  float invn = 1.0f / BN_N;
  float mean = bnsum[o] * invn;
  float var  = bnsq[o] * invn - mean * mean;      // biased variance (training mode)
  float sc   = gamma[o] * rsqrtf(var + EPSV);
  float val  = (u[id] - mean) * sc + beta[o];
  h[id] += fmaxf(val, 0.0f);
}

// ---------------- 5) global average pool over (J, T) ----------------
__global__ __launch_bounds__(256) void pool_kernel(const float* __restrict__ h,
                                                   float* __restrict__ feat) {
  int b  = blockIdx.x >> 6;
  int hc = blockIdx.x & 63;
  const float* base = h + (((size_t)b * 25) * 64 + hc) * 1024;
  float acc = 0.0f;
  for (int j = 0; j < 25; ++j) {
    const float* p = base + (size_t)j * (64 * 1024);
    for (int t = threadIdx.x; t < 1024; t += 256) acc += p[t];
  }
  __shared__ float red[256];
  red[threadIdx.x] = acc;
  __syncthreads();
  for (int s = 128; s > 0; s >>= 1) {
    if (threadIdx.x < s) red[threadIdx.x] += red[threadIdx.x + s];
    __syncthreads();
  }
  if (threadIdx.x == 0) feat[blockIdx.x] = red[0] * (1.0f / 25600.0f);
}

// ---------------- 6) classifier MLP 64->32->16->60 ----------------
__global__ __launch_bounds__(256) void mlp_kernel(const float* __restrict__ feat,
                                                  const float* __restrict__ w1, const float* __restrict__ b1,
                                                  const float* __restrict__ w2, const float* __restrict__ b2,
                                                  const float* __restrict__ w3, const float* __restrict__ b3,
                                                  float* __restrict__ out) {
  __shared__ float sf[32 * 64];
  __shared__ float s1[32 * 32];
  __shared__ float s2[32 * 16];
  int tid = threadIdx.x;
  for (int i = tid; i < 32 * 64; i += 256) sf[i] = feat[i];
  __syncthreads();
  for (int i = tid; i < 32 * 32; i += 256) {
    int b = i >> 5, o = i & 31;
    float a = b1[o];
    for (int c = 0; c < 64; ++c) a += sf[b * 64 + c] * w1[o * 64 + c];
    s1[i] = fmaxf(a, 0.0f);
  }
  __syncthreads();
  for (int i = tid; i < 32 * 16; i += 256) {
    int b = i >> 4, o = i & 15;
    float a = b2[o];
    for (int c = 0; c < 32; ++c) a += s1[b * 32 + c] * w2[o * 32 + c];
    s2[i] = fmaxf(a, 0.0f);
  }
  __syncthreads();
  for (int i = tid; i < 32 * 60; i += 256) {
    int b = i / 60, o = i - b * 60;
    float a = b3[o];
    for (int c = 0; c < 16; ++c) a += s2[b * 16 + c] * w3[o * 16 + c];
    out[i] = a;
  }
}

// ---------------- host launcher ----------------
extern "C" void kernel_launch(void* const* d_in, const int* in_sizes, int n_in,
                              void* d_out, int out_size, void* d_ws, size_t ws_size,
                              hipStream_t stream) {
  const float* x      = (const float*)d_in[0];
  const float* w_proj = (const float*)d_in[1];
  const float* b_proj = (const float*)d_in[2];
  const float* w_sp   = (const float*)d_in[3];
  const float* w_tc   = (const float*)d_in[4];
  const float* gamma  = (const float*)d_in[5];
  const float* beta   = (const float*)d_in[6];
  const float* w1     = (const float*)d_in[7];
  const float* b1     = (const float*)d_in[8];
  const float* w2     = (const float*)d_in[9];
  const float* b2     = (const float*)d_in[10];
  const float* w3     = (const float*)d_in[11];
  const float* b3     = (const float*)d_in[12];

  const size_t NH = (size_t)NELEM;          // 52,428,800 floats
  float* h     = (float*)d_ws;
  float* u     = h + NH;
  float* weff  = u + NH;                    // 3*64*64
  float* bnsum = weff + 3 * 64 * 64;        // 64
  float* bnsq  = bnsum + 64;                // 64
  float* feat  = bnsq + 64;                 // 32*64

  const unsigned eg = NELEM / 256u;         // 204800 blocks, exact

  proj_kernel<<<eg, 256, 0, stream>>>(x, w_proj, b_proj, h);

  for (int blk = 0; blk < 3; ++blk) {
    weff_kernel<<<48, 256, 0, stream>>>(w_sp, w_tc, weff, blk);
    zero_kernel<<<1, 128, 0, stream>>>(bnsum, 128);   // bnsum + bnsq contiguous
    conv_wmma_kernel<<<NROW * 8, 256, 0, stream>>>(h, weff, u, bnsum, bnsq);
    bn_apply_kernel<<<eg, 256, 0, stream>>>(u, bnsum, bnsq,
                                            gamma + blk * 64, beta + blk * 64, h);
  }

  pool_kernel<<<BAT * 64, 256, 0, stream>>>(h, feat);
  mlp_kernel<<<1, 256, 0, stream>>>(feat, w1, b1, w2, b2, w3, b3, (float*)d_out);
}